// SHEmbed_69406671503663
// MI455X (gfx1250) — compile-verified
//
#include <hip/hip_runtime.h>
#include <stdint.h>

// SH embedding: out[b, 0:3] = clamp( sum_i basis_i(dir_b) * sh_data[y_b, x_b, i, 0:3], 0, 1 )
// L = 3 -> 16 basis values; sh_data is [1024, 1024, 16, 3] f32 (48 contiguous floats per texel).
//
// Memory-bound (~0.6 FLOP/byte): the gather of 192 B/ray dominates. Strategy:
// per-lane async gather into LDS (GLOBAL_LOAD_ASYNC_TO_LDS_B128, CDNA5 ASYNCcnt path)
// overlapped with the Cartesian-polynomial SH basis evaluation, then 48 FMAs from LDS.

#define THREADS 256
#define NB 16        // basis size (L+1)^2
#define COEFS 48     // 16 * 3 floats per ray
#define WDIM 1024    // W from reference

__global__ __launch_bounds__(THREADS) void sh_embed_kernel(
    const int* __restrict__ ys,
    const int* __restrict__ xs,
    const float* __restrict__ dirs,   // [n,3]
    const float* __restrict__ sh,     // [1024,1024,16,3]
    float* __restrict__ out,          // [n,3]
    int n)
{
    __shared__ __align__(16) float smem[THREADS * COEFS];   // 48 KB / block

    const int tid = threadIdx.x;
    const int b   = blockIdx.x * THREADS + tid;
    if (b >= n) return;   // n is a multiple of 256 in practice -> EXEC stays all-ones

    const int y = ys[b];
    const int x = xs[b];

    // ---- kick off async gather: 192 B of coefficients -> this lane's LDS slot ----
    const size_t texel = (size_t)y * (size_t)WDIM + (size_t)x;
    uint64_t ga = (uint64_t)(uintptr_t)(sh + texel * COEFS);
    // low 32 bits of a generic pointer into LDS == workgroup-relative LDS byte offset
    uint32_t la = (uint32_t)(uintptr_t)(&smem[tid * COEFS]);
    #pragma unroll
    for (int i = 0; i < 12; ++i) {
        asm volatile("global_load_async_to_lds_b128 %0, %1, off"
                     :: "v"(la), "v"(ga)
                     : "memory");
        ga += 16u;
        la += 16u;
    }

    // ---- overlap: evaluate the 16 SH basis polynomials (Cartesian closed form) ----
    const float dx = dirs[3 * b + 0];
    const float dy = dirs[3 * b + 1];
    const float dz = dirs[3 * b + 2];

    const float r   = sqrtf(dx * dx + dy * dy + dz * dz) + 1e-14f;
    const float inv = 1.0f / r;
    const float xn = dx * inv, yn = dy * inv, zn = dz * inv;
    const float xx = xn * xn, yy = yn * yn, zz = zn * zn, xy = xn * yn;

    // Sign convention matches reference: columns for m<0 use sin(|m|phi), m>0 cos(m phi),
    // both scaled by sqrt2*(-1)^m * N(l,|m|) * P_l^{|m|} (CS-phase-free P, sign folded in).
    float bs[NB];
    bs[0]  =  0.28209479177387814f;
    bs[1]  = -0.48860251190291992f * yn;
    bs[2]  =  0.48860251190291992f * zn;
    bs[3]  = -0.48860251190291992f * xn;
    bs[4]  =  1.09254843059207907f * xy;
    bs[5]  = -1.09254843059207907f * yn * zn;
    bs[6]  =  0.31539156525252005f * (3.0f * zz - 1.0f);
    bs[7]  = -1.09254843059207907f * xn * zn;
    bs[8]  =  0.54627421529603953f * (xx - yy);
    bs[9]  = -0.59004358992664352f * yn * (3.0f * xx - yy);
    bs[10] =  2.89061144264055405f * xy * zn;
    bs[11] = -0.45704579946446577f * yn * (5.0f * zz - 1.0f);
    bs[12] =  0.37317633259011546f * zn * (5.0f * zz - 3.0f);
    bs[13] = -0.45704579946446577f * xn * (5.0f * zz - 1.0f);
    bs[14] =  1.44530572132027735f * zn * (xx - yy);
    bs[15] = -0.59004358992664352f * xn * (xx - 3.0f * yy);

    // ---- wait for our async gather (per-wave ASYNCcnt), then contract from LDS ----
    asm volatile("s_wait_asynccnt 0" ::: "memory");

    const float4* c4 = reinterpret_cast<const float4*>(&smem[tid * COEFS]);
    float c[COEFS];
    #pragma unroll
    for (int i = 0; i < 12; ++i) {
        const float4 v = c4[i];
        c[4 * i + 0] = v.x;
        c[4 * i + 1] = v.y;
        c[4 * i + 2] = v.z;
        c[4 * i + 3] = v.w;
    }

    float r0 = 0.0f, r1 = 0.0f, r2 = 0.0f;
    #pragma unroll
    for (int i = 0; i < NB; ++i) {
        r0 = fmaf(bs[i], c[3 * i + 0], r0);
        r1 = fmaf(bs[i], c[3 * i + 1], r1);
        r2 = fmaf(bs[i], c[3 * i + 2], r2);
    }

    r0 = fminf(fmaxf(r0, 0.0f), 1.0f);
    r1 = fminf(fmaxf(r1, 0.0f), 1.0f);
    r2 = fminf(fmaxf(r2, 0.0f), 1.0f);

    out[3 * b + 0] = r0;
    out[3 * b + 1] = r1;
    out[3 * b + 2] = r2;
}

extern "C" void kernel_launch(void* const* d_in, const int* in_sizes, int n_in,
                              void* d_out, int out_size, void* d_ws, size_t ws_size,
                              hipStream_t stream) {
    (void)n_in; (void)out_size; (void)d_ws; (void)ws_size;
    const int*   ys   = (const int*)d_in[0];
    const int*   xs   = (const int*)d_in[1];
    const float* dirs = (const float*)d_in[2];
    const float* sh   = (const float*)d_in[3];
    float*       out  = (float*)d_out;

    const int n = in_sizes[0];
    const int grid = (n + THREADS - 1) / THREADS;
    sh_embed_kernel<<<grid, THREADS, 0, stream>>>(ys, xs, dirs, sh, out, n);
}